// CenterLoss_1357209665637
// MI455X (gfx1250) — compile-verified
//
#include <hip/hip_runtime.h>

typedef float v2f __attribute__((ext_vector_type(2)));
typedef float v8f __attribute__((ext_vector_type(8)));

#define B_DIM 8192
#define C_DIM 10000
#define D_DIM 128
#define WAVES_PER_BLOCK 4
#define ROWS_PER_WAVE 16
#define NUM_BLOCKS (B_DIM / (WAVES_PER_BLOCK * ROWS_PER_WAVE))   // 128

// Each wave computes, for its 16 rows b, dist_b = ||x_b||^2 + ||c_lab||^2 - 2<x_b,c_lab>
// using chained V_WMMA_F32_16X16X4_F32 (diagonal of the 16x16 tile), clamps, and
// reduces. One deterministic partial per block goes to d_ws.
__global__ void __launch_bounds__(WAVES_PER_BLOCK * 32)
center_loss_main(const float* __restrict__ x,
                 const int* __restrict__ labels,
                 const float* __restrict__ centers,
                 float* __restrict__ partials) {
    const int lane = threadIdx.x & 31;
    const int wave = threadIdx.x >> 5;
    const int b0   = (blockIdx.x * WAVES_PER_BLOCK + wave) * ROWS_PER_WAVE;

    const int row = lane & 15;   // M index for A frag, N index for B frag
    const int hi  = lane >> 4;   // selects K-half {0,1} vs {2,3} per ISA layout

    const int   b   = b0 + row;
    const int   lab = labels[b];

    const float* xp = x       + (size_t)b   * D_DIM + 2 * hi;
    const float* cp = centers + (size_t)lab * D_DIM + 2 * hi;

    v8f acc = {};
    float sx = 0.0f;   // partial sum of squares of x elements this lane touches
    float sc = 0.0f;   // same for the gathered center row

    // A: 16x4 f32 frag = v2f per lane; B: 4x16 f32 frag = v2f per lane.
    // 32 chained K=4 steps cover D=128.
    #pragma unroll 4
    for (int k = 0; k < D_DIM; k += 4) {
        v2f a  = *(const v2f*)(xp + k);
        v2f bf = *(const v2f*)(cp + k);
        sx += a.x * a.x + a.y * a.y;
        sc += bf.x * bf.x + bf.y * bf.y;
        acc = __builtin_amdgcn_wmma_f32_16x16x4_f32(
            /*neg_a=*/false, a, /*neg_b=*/false, bf,
            /*c_mod=*/(short)0, acc, /*reuse_a=*/false, /*reuse_b=*/false);
    }

    // Combine the two K-halves so every lane has the full row norms for row (lane&15).
    sx += __shfl_xor(sx, 16, 32);
    sc += __shfl_xor(sc, 16, 32);

    // Diagonal of 16x16 f32 C/D layout: lanes 0..7 hold diag[m]=acc[lane];
    // lanes 24..31 hold diag[lane-16]=acc[lane-24]; lanes 8..23 hold none.
    const bool has = (lane < 8) || (lane >= 24);
    const int  di  = (lane < 8) ? lane : (lane - 24);
    float dot = 0.0f;
    #pragma unroll
    for (int i = 0; i < 8; ++i) dot = (di == i) ? acc[i] : dot;

    float dist = sx + sc - 2.0f * dot;
    dist = fminf(fmaxf(dist, 1e-12f), 1e12f);
    float contrib = has ? dist : 0.0f;

    // Fixed-order wave32 reduction.
    #pragma unroll
    for (int off = 16; off > 0; off >>= 1)
        contrib += __shfl_down(contrib, off, 32);

    __shared__ float wsum[WAVES_PER_BLOCK];
    if (lane == 0) wsum[wave] = contrib;
    __syncthreads();
    if (threadIdx.x == 0)
        partials[blockIdx.x] = (wsum[0] + wsum[1]) + (wsum[2] + wsum[3]);
}

// Deterministic fixed-order tree reduction of the 128 block partials,
// then loss = sum/B + (C-1)*1e-12 (the clamped zeros of the masked columns).
__global__ void __launch_bounds__(NUM_BLOCKS)
center_loss_finalize(const float* __restrict__ partials, float* __restrict__ out) {
    __shared__ float sm[NUM_BLOCKS];
    const int t = threadIdx.x;
    sm[t] = partials[t];
    __syncthreads();
    #pragma unroll
    for (int s = NUM_BLOCKS / 2; s > 0; s >>= 1) {
        if (t < s) sm[t] += sm[t + s];
        __syncthreads();
    }
    if (t == 0)
        out[0] = sm[0] / (float)B_DIM + (float)(C_DIM - 1) * 1e-12f;
}

extern "C" void kernel_launch(void* const* d_in, const int* in_sizes, int n_in,
                              void* d_out, int out_size, void* d_ws, size_t ws_size,
                              hipStream_t stream) {
    const float* x       = (const float*)d_in[0];
    const int*   labels  = (const int*)d_in[1];
    const float* centers = (const float*)d_in[2];
    float* partials = (float*)d_ws;   // NUM_BLOCKS floats of scratch

    center_loss_main<<<NUM_BLOCKS, WAVES_PER_BLOCK * 32, 0, stream>>>(
        x, labels, centers, partials);
    center_loss_finalize<<<1, NUM_BLOCKS, 0, stream>>>(partials, (float*)d_out);
}